// HybridTransformerClassifier_65481071396476
// MI455X (gfx1250) — compile-verified
//
#include <hip/hip_runtime.h>

typedef __attribute__((ext_vector_type(2))) float v2f;
typedef __attribute__((ext_vector_type(8))) float v8f;

#define SEQ   2048
#define DIM   256
#define BATCH 32
#define NCLS  10
#define NPAD  16
#define INV_2PI 0.15915494309189535f

// ---------------------------------------------------------------------------
// Kernel 0: zero the pooled accumulator in workspace (harness poisons d_ws).
// ---------------------------------------------------------------------------
__global__ void zero_ws_kernel(float* __restrict__ p, int n) {
    int i = blockIdx.x * blockDim.x + threadIdx.x;
    if (i < n) p[i] = 0.0f;
}

// ---------------------------------------------------------------------------
// Kernel 1: gather + pos add + 4x cumprod(cos(.)) scan + mean-pool.
// One wave32 per (b,s) row: lane holds 8 consecutive D-elements.
// 8 waves per block -> all 8 rows share the same batch index b.
// ---------------------------------------------------------------------------
__global__ void embed_gate_pool_kernel(const int*   __restrict__ x,
                                       const float* __restrict__ emb,
                                       const float* __restrict__ pos,
                                       float*       __restrict__ pooled) {
    __shared__ float lds[DIM];

    const int tid  = threadIdx.x;
    const int lane = tid & 31;
    const int wave = tid >> 5;                 // 0..7
    const int row  = blockIdx.x * 8 + wave;    // [0, 32*2048)
    const int b    = row >> 11;                // row / 2048
    const int s    = row & (SEQ - 1);          // row % 2048

    if (tid < DIM) lds[tid] = 0.0f;
    __syncthreads();

    // Gather embedding row + positional row: 2x float4 per lane.
    const int tok = x[row];
    const float4* erow = (const float4*)(emb + (size_t)tok * DIM);
    const float4* prow = (const float4*)(pos + (size_t)s * DIM);
    float4 e0 = erow[lane * 2 + 0];
    float4 e1 = erow[lane * 2 + 1];
    float4 p0 = prow[lane * 2 + 0];
    float4 p1 = prow[lane * 2 + 1];

    float h[8];
    h[0] = e0.x + p0.x;  h[1] = e0.y + p0.y;
    h[2] = e0.z + p0.z;  h[3] = e0.w + p0.w;
    h[4] = e1.x + p1.x;  h[5] = e1.y + p1.y;
    h[6] = e1.z + p1.z;  h[7] = e1.w + p1.w;

    // 4 quantum-gate blocks: h = cumprod(cos(h)) along the 256-wide row.
    #pragma unroll
    for (int blk = 0; blk < 4; ++blk) {
        // per-lane serial cumulative product of cos over 8 elements
        float p = 1.0f;
        #pragma unroll
        for (int j = 0; j < 8; ++j) {
            float c = __builtin_amdgcn_cosf(h[j] * INV_2PI);  // v_cos_f32
            p *= c;
            h[j] = p;
        }
        // multiplicative inclusive scan of lane totals across the wave (5 steps)
        float v = p;
        #pragma unroll
        for (int off = 1; off < 32; off <<= 1) {
            float t = __shfl_up(v, (unsigned)off, 32);
            v *= (lane >= off) ? t : 1.0f;
        }
        // exclusive prefix for this lane
        float excl = __shfl_up(v, 1u, 32);
        excl = (lane == 0) ? 1.0f : excl;
        #pragma unroll
        for (int j = 0; j < 8; ++j) h[j] *= excl;
    }

    // Block-level reduction over the 8 rows (same b) into LDS (ds_add_f32),
    // then one global fp atomic per (b, d) per block, pre-scaled for the mean.
    #pragma unroll
    for (int j = 0; j < 8; ++j)
        atomicAdd(&lds[lane * 8 + j], h[j]);
    __syncthreads();

    if (tid < DIM)
        atomicAdd(&pooled[b * DIM + tid], lds[tid] * (1.0f / (float)SEQ));
}

// ---------------------------------------------------------------------------
// Kernel 2: pooled[32,256] @ W[256,10] + b via V_WMMA_F32_16X16X4_F32.
// 1 block, 2 waves; each wave owns one 16x16 output tile (N padded 10->16).
// W is staged zero-padded into LDS [256][16] once, so the K-loop has NO
// divergence (EXEC stays all-ones, as WMMA requires) and B comes from two
// conflict-free ds_load_b32 per iteration. K=256 -> 64 f32 WMMA per wave.
//
// 16x4 f32 A layout (ISA 7.12.2): VGPR0 = K0 (lanes 0-15) / K2 (lanes 16-31),
// VGPR1 = K1 / K3. B (4x16) mirrors with rows striped across lanes.
// ---------------------------------------------------------------------------
__global__ void final_gemm_wmma_kernel(const float* __restrict__ P,
                                       const float* __restrict__ W,
                                       const float* __restrict__ bias,
                                       float*       __restrict__ out) {
    __shared__ float Wp[DIM * NPAD];   // [256][16], cols 10..15 zero
    __shared__ float bp[NPAD];

    const int tid  = threadIdx.x;      // 0..63
    const int lane = tid & 31;
    const int wave = tid >> 5;         // 0 or 1 -> M tile
    const int m0   = wave * 16;
    const int half = lane >> 4;        // 0: lanes 0-15, 1: lanes 16-31
    const int l16  = lane & 15;

    // Stage padded W + bias into LDS (uniform loop, no divergence).
    for (int i = tid; i < DIM * NPAD; i += 64) {
        const int k = i >> 4;
        const int c = i & (NPAD - 1);
        Wp[i] = (c < NCLS) ? W[k * NCLS + c] : 0.0f;
    }
    if (tid < NPAD) bp[tid] = (tid < NCLS) ? bias[tid] : 0.0f;
    __syncthreads();

    const int col  = l16;              // output column this lane covers
    const int arow = m0 + l16;

    v8f acc = {};
    for (int k0 = 0; k0 < DIM; k0 += 4) {
        const int kb = k0 + half * 2;

        v2f a;                         // A: one global_load_b64, contiguous
        a.x = P[arow * DIM + kb + 0];
        a.y = P[arow * DIM + kb + 1];

        v2f bm;                        // B: two bank-conflict-free ds_load_b32
        bm.x = Wp[(kb + 0) * NPAD + col];
        bm.y = Wp[(kb + 1) * NPAD + col];

        acc = __builtin_amdgcn_wmma_f32_16x16x4_f32(
            /*neg_a=*/false, a, /*neg_b=*/false, bm,
            /*c_mod=*/(short)0, acc, /*reuse_a=*/false, /*reuse_b=*/false);
    }

    // C/D layout: VGPR j -> row (m0 + half*8 + j), col = l16.
    const float bv = bp[col];
    #pragma unroll
    for (int j = 0; j < 8; ++j) {
        const int r = m0 + half * 8 + j;
        if (col < NCLS) out[r * NCLS + col] = acc[j] + bv;
    }
}

// ---------------------------------------------------------------------------
extern "C" void kernel_launch(void* const* d_in, const int* in_sizes, int n_in,
                              void* d_out, int out_size, void* d_ws, size_t ws_size,
                              hipStream_t stream) {
    const int*   x    = (const int*)  d_in[0];   // [32, 2048] int32
    const float* emb  = (const float*)d_in[1];   // [50257, 256]
    const float* pos  = (const float*)d_in[2];   // [1, 5000, 256]
    const float* W    = (const float*)d_in[3];   // [256, 10]
    const float* bias = (const float*)d_in[4];   // [10]
    float* out    = (float*)d_out;               // [32, 10]
    float* pooled = (float*)d_ws;                // [32, 256] accumulator

    const int pooled_n = BATCH * DIM;            // 8192 floats
    zero_ws_kernel<<<(pooled_n + 255) / 256, 256, 0, stream>>>(pooled, pooled_n);

    const int rows = BATCH * SEQ;                // 65536 rows, 8 per block
    embed_gate_pool_kernel<<<rows / 8, 256, 0, stream>>>(x, emb, pos, pooled);

    final_gemm_wmma_kernel<<<1, 64, 0, stream>>>(pooled, W, bias, out);
}